// RelationGNN_40716289966339
// MI455X (gfx1250) — compile-verified
//
#include <hip/hip_runtime.h>

// GraphSAGE-mean on MI455X (gfx1250, wave32).
//
// Roofline: edge phase moves ~660 MB (gather + f32 atomic scatter) -> memory
// bound at 23.3 TB/s. GEMM phase is 2 x [100000x128] @ [128x128] = 6.55 GFLOP
// over ~100 MB I/O -> trivially compute-light, so we use the native-precision
// V_WMMA_F32_16X16X4_F32 path (full f32, matches reference numerics).

#define N_USERS   100000
#define EMBED_DIM 128
#define N_EDGES   640000

typedef __attribute__((ext_vector_type(2))) float v2f;
typedef __attribute__((ext_vector_type(8))) float v8f;

// ---------------------------------------------------------------------------
// Phase 1: one wave per edge. 32 lanes x float4 = 128 floats gathered from
// emb[src], scattered with hardware global_atomic_add_f32 into agg[dst]
// (agg lives in d_out). Lane 0 bumps the degree counter.
// ---------------------------------------------------------------------------
__global__ void __launch_bounds__(256)
sage_scatter(const float* __restrict__ emb,
             const int*   __restrict__ src,
             const int*   __restrict__ dst,
             float*       __restrict__ agg,
             float*       __restrict__ deg) {
    const int edge = blockIdx.x * 8 + (threadIdx.x >> 5);
    if (edge >= N_EDGES) return;
    const int lane = threadIdx.x & 31;
    const int s = src[edge];
    const int d = dst[edge];

    const float4 v =
        reinterpret_cast<const float4*>(emb + (size_t)s * EMBED_DIM)[lane];
    float* ap = agg + (size_t)d * EMBED_DIM + lane * 4;

    __hip_atomic_fetch_add(ap + 0, v.x, __ATOMIC_RELAXED, __HIP_MEMORY_SCOPE_AGENT);
    __hip_atomic_fetch_add(ap + 1, v.y, __ATOMIC_RELAXED, __HIP_MEMORY_SCOPE_AGENT);
    __hip_atomic_fetch_add(ap + 2, v.z, __ATOMIC_RELAXED, __HIP_MEMORY_SCOPE_AGENT);
    __hip_atomic_fetch_add(ap + 3, v.w, __ATOMIC_RELAXED, __HIP_MEMORY_SCOPE_AGENT);
    if (lane == 0)
        __hip_atomic_fetch_add(deg + d, 1.0f, __ATOMIC_RELAXED, __HIP_MEMORY_SCOPE_AGENT);
}

// ---------------------------------------------------------------------------
// Phase 2: per block of 16 rows: normalize agg (=d_out rows) by degree into
// LDS, stage emb rows into LDS, then 8 waves each compute a 16x16 output tile
// via chained V_WMMA_F32_16X16X4_F32:
//     C += A_agg x W_l^T   and   C += A_emb x W_r^T
// B-fragment trick: B[k][n] = W[n][k], and the f32 WMMA B layout wants
// lane (n = lane&15) holding K = k + 2*(lane>>4), k+1+2*(lane>>4) -> that is a
// contiguous float2 from row (n0 + lane&15) of W. No transpose needed.
// ---------------------------------------------------------------------------
#define LDS_STRIDE 132   // 528 B rows: float4-aligned, bank-conflict-free cols

__global__ void __launch_bounds__(256)
sage_gemm(const float* __restrict__ emb,
          const float* __restrict__ W_l,
          const float* __restrict__ b_l,
          const float* __restrict__ W_r,
          const float* __restrict__ deg,
          float*       __restrict__ inout /* agg in, out out (same rows) */) {
    __shared__ float s_agg[16 * LDS_STRIDE];
    __shared__ float s_emb[16 * LDS_STRIDE];

    const int m0 = blockIdx.x * 16;
    const int t  = threadIdx.x;

    // Stage tiles: thread t loads 8 floats of row (t>>4), cols (t&15)*8 ..
    {
        const int   r   = t >> 4;
        const int   cb  = (t & 15) * 8;
        const float inv = 1.0f / fmaxf(deg[m0 + r], 1.0f);
        const float4* ga =
            reinterpret_cast<const float4*>(inout + (size_t)(m0 + r) * EMBED_DIM + cb);
        const float4* ge =
            reinterpret_cast<const float4*>(emb + (size_t)(m0 + r) * EMBED_DIM + cb);
        const float4 a0 = ga[0], a1 = ga[1];
        const float4 e0 = ge[0], e1 = ge[1];
        float4* la = reinterpret_cast<float4*>(s_agg + r * LDS_STRIDE + cb);
        float4* le = reinterpret_cast<float4*>(s_emb + r * LDS_STRIDE + cb);
        la[0] = make_float4(a0.x * inv, a0.y * inv, a0.z * inv, a0.w * inv);
        la[1] = make_float4(a1.x * inv, a1.y * inv, a1.z * inv, a1.w * inv);
        le[0] = e0;
        le[1] = e1;
    }
    __syncthreads();

    const int lane  = t & 31;
    const int wave  = t >> 5;
    const int n0    = wave * 16;           // this wave's 16-column slab
    const int rsel  = lane & 15;           // row within fragment / col of B
    const int khalf = (lane >> 4) * 2;     // K sub-offset per A/B f32 layout

    const float* pa = s_agg + rsel * LDS_STRIDE + khalf;
    const float* pe = s_emb + rsel * LDS_STRIDE + khalf;
    const float* wl = W_l + (size_t)(n0 + rsel) * EMBED_DIM + khalf;
    const float* wr = W_r + (size_t)(n0 + rsel) * EMBED_DIM + khalf;

    v8f c = {};
#pragma unroll
    for (int kk = 0; kk < EMBED_DIM; kk += 4) {
        const v2f aA = *reinterpret_cast<const v2f*>(pa + kk);
        const v2f aE = *reinterpret_cast<const v2f*>(pe + kk);
        const v2f bL = *reinterpret_cast<const v2f*>(wl + kk);
        const v2f bR = *reinterpret_cast<const v2f*>(wr + kk);
        // 8 args: (neg_a, A, neg_b, B, c_mod, C, reuse_a, reuse_b)
        c = __builtin_amdgcn_wmma_f32_16x16x4_f32(false, aA, false, bL,
                                                  (short)0, c, false, false);
        c = __builtin_amdgcn_wmma_f32_16x16x4_f32(false, aE, false, bR,
                                                  (short)0, c, false, false);
    }

    // C/D layout: lane holds col n0+(lane&15); VGPR r holds row
    // m0 + r + 8*(lane>>4). Bias b_l is per-column.
    const float bias  = b_l[n0 + rsel];
    const int   col   = n0 + rsel;
    const int   mbase = m0 + (lane >> 4) * 8;
#pragma unroll
    for (int r = 0; r < 8; ++r)
        inout[(size_t)(mbase + r) * EMBED_DIM + col] = c[r] + bias;
}

// ---------------------------------------------------------------------------
extern "C" void kernel_launch(void* const* d_in, const int* in_sizes, int n_in,
                              void* d_out, int out_size, void* d_ws, size_t ws_size,
                              hipStream_t stream) {
    const float* emb = (const float*)d_in[0];
    const float* W_l = (const float*)d_in[1];
    const float* b_l = (const float*)d_in[2];
    const float* W_r = (const float*)d_in[3];
    const int*   src = (const int*)d_in[4];
    const int*   dst = (const int*)d_in[5];

    float* out = (float*)d_out;   // doubles as agg accumulator, then output
    float* deg = (float*)d_ws;    // N_USERS floats of scratch

    // Re-zero accumulators every call (graph replays must be deterministic).
    hipMemsetAsync(out, 0, (size_t)N_USERS * EMBED_DIM * sizeof(float), stream);
    hipMemsetAsync(deg, 0, (size_t)N_USERS * sizeof(float), stream);

    sage_scatter<<<N_EDGES / 8, 256, 0, stream>>>(emb, src, dst, out, deg);
    sage_gemm<<<N_USERS / 16, 256, 0, stream>>>(emb, W_l, b_l, W_r, deg, out);
}